// cross_correlation_12249246728709
// MI455X (gfx1250) — compile-verified
//
#include <hip/hip_runtime.h>
#include <hip/hip_bf16.h>

// Fused: Linear(L,L) x2 -> full cross-correlation (depthwise) -> Conv1d(2L->L,k=3)
// One workgroup (128 threads = 4 wave32) per (batch b, 16-wide d tile).
// Linears + final conv use V_WMMA_F32_16X16X4_F32 (full fp32 precision).
// Stage A input staging uses the Tensor Data Mover (tensor_load_to_lds) where
// the tile is representable (all tiles except the left-edge halo tile).
// The depthwise correlation uses a register sliding-window VALU loop.

typedef float v2f __attribute__((ext_vector_type(2)));
typedef float v8f __attribute__((ext_vector_type(8)));
typedef unsigned int u32x4 __attribute__((ext_vector_type(4)));
typedef int i32x4 __attribute__((ext_vector_type(4)));
typedef int i32x8 __attribute__((ext_vector_type(8)));

namespace {
constexpr int kB   = 256;
constexpr int kL   = 64;    // length
constexpr int kD   = 1024;  // channels
constexpr int kTL  = 128;   // 2L correlation lags
constexpr int kNC  = 18;    // 16 core columns + 1 halo each side (conv k=3 along d)
constexpr int kStr = 18;    // LDS row stride (floats); gcd(18,64)=2 -> conflict-free
}

#define XCORR_HAS_TDM __has_builtin(__builtin_amdgcn_tensor_load_to_lds)

#if XCORR_HAS_TDM
// Issue one TDM 2D tile load: 64 rows x 18 cols of f32, row stride kD elems.
// tdim0 = elements remaining from tile start (right-edge OOB zero-fills).
__device__ __forceinline__ void tdm_load_tile(const float* gsrc, unsigned lds_byte,
                                              unsigned tdim0) {
  const unsigned long long ga = (unsigned long long)(uintptr_t)gsrc;
  u32x4 g0;
  g0[0] = 1u;                                   // count=1, no gather, user desc
  g0[1] = lds_byte;                             // D#.lds_addr
  g0[2] = (unsigned)ga;                         // global_addr[31:0]
  g0[3] = ((unsigned)(ga >> 32) & 0x01FFFFFFu)  // global_addr[56:32]
        | (2u << 30);                           // type = 2 ("image")
  i32x8 g1;
  g1[0] = (int)(2u << 16);                      // wg_mask=0, data_size=2 (4B)
  g1[1] = (int)((tdim0 & 0xFFFFu) << 16);       // tensor_dim0[15:0]
  g1[2] = (int)((tdim0 >> 16) | ((unsigned)kL << 16)); // tensor_dim0[31:16] | tensor_dim1=64
  g1[3] = (int)((unsigned)kNC << 16);           // tensor_dim1[31:16]=0 | tile_dim0=18
  g1[4] = (int)kL;                              // tile_dim1=64, tile_dim2=0
  g1[5] = (int)kD;                              // tensor_dim0_stride[31:0] = 1024
  g1[6] = 0;                                    // stride hi | tensor_dim1_stride lo
  g1[7] = 0;
  i32x4 gz4 = {0, 0, 0, 0};
#if defined(__clang_major__) && (__clang_major__ >= 23)
  i32x8 gz8 = {0, 0, 0, 0, 0, 0, 0, 0};
  __builtin_amdgcn_tensor_load_to_lds(g0, g1, gz4, gz4, gz8, 0);
#else
  __builtin_amdgcn_tensor_load_to_lds(g0, g1, gz4, gz4, 0);
#endif
}
#endif

__global__ __launch_bounds__(128) void fused_xcorr_conv(
    const float* __restrict__ left, const float* __restrict__ right,
    const float* __restrict__ Wl,   const float* __restrict__ bl,
    const float* __restrict__ Wr,   const float* __restrict__ br,
    const float* __restrict__ Wconv, float* __restrict__ y)
{
  __shared__ float sL [kL  * kStr];   // raw left slab  [l][c]
  __shared__ float sR [kL  * kStr];   // raw right slab
  __shared__ float sLL[kL  * kStr];   // Wl @ left + bl
  __shared__ float sRR[kL  * kStr];   // Wr @ right + br
  __shared__ float sC [kTL * kStr];   // correlation [i=0..127][c]

  const int tid  = threadIdx.x;
  const int lane = tid & 31;
  const int wv   = tid >> 5;
  const int b    = blockIdx.x >> 6;   // 64 d-tiles per batch
  const int tile = blockIdx.x & 63;
  const int d0   = tile << 4;

  // ---------------- Stage A: stage raw slabs (with d halo) into LDS ----------
  auto stageA_manual = [&]() {
    for (int idx = tid; idx < kL * kNC; idx += 128) { // 1152 = 9*128, uniform
      const int row = idx / kNC;
      const int c   = idx - row * kNC;
      const int d   = d0 - 1 + c;
      float lv = 0.f, rv = 0.f;
      if ((unsigned)d < (unsigned)kD) {
        const size_t g = ((size_t)b * kL + row) * kD + d;
        lv = left[g];
        rv = right[g];
      }
      sL[row * kStr + c] = lv;
      sR[row * kStr + c] = rv;
    }
  };

#if XCORR_HAS_TDM
  if (tile != 0) {
    // Tile spans global columns [d0-1, d0+17); all in-range on the left,
    // right edge zero-filled by TDM OOB handling via tensor_dim0.
    if (wv == 0) {
      const size_t elem0 = ((size_t)b * kL) * kD + (size_t)(d0 - 1);
      const unsigned tdim0 = (unsigned)(kD - (d0 - 1)); // 1025 - d0
      tdm_load_tile(left  + elem0, (unsigned)(uintptr_t)sL, tdim0);
      tdm_load_tile(right + elem0, (unsigned)(uintptr_t)sR, tdim0);
    }
    __builtin_amdgcn_s_wait_tensorcnt(0);  // no-op for waves with TENSORcnt==0
  } else {
    stageA_manual();                       // left halo column is d = -1
  }
#else
  stageA_manual();
#endif
  __syncthreads();

  // WMMA fragment geometry (wave32, 16x16x4 f32)
  const int m0   = wv << 4;           // wave's M tile (output rows)
  const int half = lane >> 4;         // 0 | 1
  const int mrow = m0 + (lane & 15);  // A-matrix row held by this lane
  const int ksub = half << 1;         // A/B hold K = ks*4 + {0,1} | {2,3}
  const int ncol = lane & 15;         // B/C/D column
  const int hi8  = half << 3;         // C/D rows m0+r | m0+8+r

  // ---------------- Stage B: the two linears via WMMA -----------------------
  // llin[k,c] = sum_l Wl[k,l] * x[l,c] + bl[k]; 18 cols via tiles at cb=0 and 2
  for (int which = 0; which < 2; ++which) {
    const float* __restrict__ W    = which ? Wr : Wl;
    const float* __restrict__ bias = which ? br : bl;
    const float* __restrict__ src  = which ? sR  : sL;
    float*       __restrict__ dst  = which ? sRR : sLL;
    for (int cb = 0; cb <= 2; cb += 2) {
      v8f acc;
#pragma unroll
      for (int r = 0; r < 8; ++r) acc[r] = bias[m0 + hi8 + r];
#pragma unroll
      for (int ks = 0; ks < 16; ++ks) {            // K = 64, 4 per WMMA
        const int kb = ks * 4 + ksub;
        v2f a, bf;
        a.x  = W[mrow * kL + kb];
        a.y  = W[mrow * kL + kb + 1];
        bf.x = src[kb       * kStr + cb + ncol];
        bf.y = src[(kb + 1) * kStr + cb + ncol];
        acc = __builtin_amdgcn_wmma_f32_16x16x4_f32(
            false, a, false, bf, (short)0, acc, false, false);
      }
#pragma unroll
      for (int r = 0; r < 8; ++r)
        dst[(m0 + hi8 + r) * kStr + cb + ncol] = acc[r];
    }
  }
  __syncthreads();

  // ---------------- Stage C: depthwise cross-correlation (VALU) -------------
  // corr[i,c] = sum_j llin[j,c] * rlin[j+63-i,c]   (out-of-range -> 0)
  // Core 16 columns: thread = (c, 16-lag block); 16-deep register window of
  // rlin slides with j: 2 LDS loads per 16 FMAs.
  {
    const int c  = (tid & 15) + 1;    // LDS cols 1..16 == global d0..d0+15
    const int i0 = (tid >> 4) << 4;   // lag block [i0, i0+16)
    float w[16], acc[16];
#pragma unroll
    for (int k = 0; k < 16; ++k) {
      acc[k] = 0.f;
      const int row = 62 - i0 - k;    // window state at j = -1
      w[k] = ((unsigned)row < (unsigned)kL) ? sRR[row * kStr + c] : 0.f;
    }
#pragma unroll
    for (int j = 0; j < kL; ++j) {
      const float lv = sLL[j * kStr + c];
#pragma unroll
      for (int k = 15; k > 0; --k) w[k] = w[k - 1];   // register renaming
      const int rn = j + 63 - i0;
      w[0] = ((unsigned)rn < (unsigned)kL) ? sRR[rn * kStr + c] : 0.f;
#pragma unroll
      for (int k = 0; k < 16; ++k) acc[k] += lv * w[k];
    }
#pragma unroll
    for (int k = 0; k < 16; ++k) sC[(i0 + k) * kStr + c] = acc[k];
  }
  // Halo columns (LDS col 0 and 17): 2-lag blocks, all 128 threads (~1/8 cost).
  // Conv1d zero-pads along d, so out-of-range global columns store 0.
  {
    const int hc     = (tid & 1) ? 17 : 0;
    const int i0     = (tid >> 1) << 1;           // lag block [i0, i0+2)
    const int dglob  = d0 - 1 + hc;
    const bool dvalid = (unsigned)dglob < (unsigned)kD;
    float w0, w1, a0 = 0.f, a1 = 0.f;
    {
      const int r0 = 62 - i0, r1 = 61 - i0;
      w0 = ((unsigned)r0 < (unsigned)kL) ? sRR[r0 * kStr + hc] : 0.f;
      w1 = ((unsigned)r1 < (unsigned)kL) ? sRR[r1 * kStr + hc] : 0.f;
    }
#pragma unroll
    for (int j = 0; j < kL; ++j) {
      const float lv = sLL[j * kStr + hc];
      w1 = w0;
      const int rn = j + 63 - i0;
      w0 = ((unsigned)rn < (unsigned)kL) ? sRR[rn * kStr + hc] : 0.f;
      a0 += lv * w0;
      a1 += lv * w1;
    }
    sC[ i0      * kStr + hc] = dvalid ? a0 : 0.f;
    sC[(i0 + 1) * kStr + hc] = dvalid ? a1 : 0.f;
  }
  __syncthreads();

  // ---------------- Stage D: Conv1d(2L->L, k=3) via WMMA --------------------
  // y[k, d0+n] = sum_t sum_i Wconv[k,i,t] * corr[i, n+t]  (LDS col = n+t)
  v8f yacc;
#pragma unroll
  for (int r = 0; r < 8; ++r) yacc[r] = 0.f;
#pragma unroll
  for (int t = 0; t < 3; ++t) {
#pragma unroll
    for (int ks = 0; ks < 32; ++ks) {              // K = 128, 4 per WMMA
      const int ib = ks * 4 + ksub;
      v2f a, bf;
      a.x  = Wconv[(mrow * kTL + ib)     * 3 + t];
      a.y  = Wconv[(mrow * kTL + ib + 1) * 3 + t];
      bf.x = sC[ ib      * kStr + ncol + t];
      bf.y = sC[(ib + 1) * kStr + ncol + t];
      yacc = __builtin_amdgcn_wmma_f32_16x16x4_f32(
          false, a, false, bf, (short)0, yacc, false, false);
    }
  }
#pragma unroll
  for (int r = 0; r < 8; ++r) {
    const int row = m0 + hi8 + r;
    y[((size_t)b * kL + row) * kD + d0 + ncol] = yacc[r];
  }
}

extern "C" void kernel_launch(void* const* d_in, const int* in_sizes, int n_in,
                              void* d_out, int out_size, void* d_ws, size_t ws_size,
                              hipStream_t stream) {
  (void)in_sizes; (void)n_in; (void)d_ws; (void)ws_size; (void)out_size;
  const float* left  = (const float*)d_in[0];
  const float* right = (const float*)d_in[1];
  const float* Wl    = (const float*)d_in[2];
  const float* bl    = (const float*)d_in[3];
  const float* Wr    = (const float*)d_in[4];
  const float* br    = (const float*)d_in[5];
  const float* Wconv = (const float*)d_in[6];
  float* y = (float*)d_out;

  dim3 grid(kB * (kD / 16));   // 16384 workgroups: (b, d-tile)
  dim3 block(128);             // 4 wave32
  fused_xcorr_conv<<<grid, block, 0, stream>>>(left, right, Wl, bl, Wr, br,
                                               Wconv, y);
}